// GraphConvolutionLayer_78219944394958
// MI455X (gfx1250) — compile-verified
//
#include <hip/hip_runtime.h>

typedef float v2f __attribute__((ext_vector_type(2)));
typedef float v8f __attribute__((ext_vector_type(8)));

#define DIN 256
#define DOUT 64

// ---------------------------------------------------------------------------
// Kernel 1: support = X @ W  via V_WMMA_F32_16X16X4_F32 (fp32-exact matrix op)
// One wave computes a 16x64 output tile (4 v8f accumulators), K-loop step 4.
//
// W (256x64 fp32 = 64 KB) is staged in LDS *pre-swizzled into the WMMA
// B-fragment layout*: for k-group kg (K = 4*kg..4*kg+3) and n-tile n,
//   WsB[(kg*4 + n)*64 + lane*2 + v] = W[4*kg + 2*(lane>>4) + v][n*16 + (lane&15)]
// so each lane's B fragment is one contiguous, 8B-aligned float2 ->
// a single ds_load_b64 (4 n-tiles combine into ds_load_2addr_b64), with no
// register repacking and no LDS bank conflicts.
// ---------------------------------------------------------------------------
__global__ void __launch_bounds__(256)
gcn_gemm_wmma(const float* __restrict__ X, const float* __restrict__ W,
              float* __restrict__ support, int n_nodes, int n_tiles) {
    __shared__ __align__(16) float WsB[DIN * DOUT];  // 64 KB, fragment-ordered

    // Cooperative swizzle-stage: coalesced global reads, scattered LDS writes.
    for (int i = threadIdx.x; i < DIN * DOUT; i += blockDim.x) {
        int k  = i >> 6;              // W row (K index), 0..255
        int c  = i & 63;              // W col (N index), 0..63
        int n  = c >> 4;              // n-tile
        int r  = c & 15;              // N within tile
        int kg = k >> 2;              // k-group
        int kr = k & 3;               // position within k-group = 2*half + v
        int half = kr >> 1;
        int v    = kr & 1;
        int lane = half * 16 + r;
        WsB[((kg * 4 + n) << 6) + lane * 2 + v] = W[i];
    }
    __syncthreads();

    const int wave = threadIdx.x >> 5;   // 0..7
    const int lane = threadIdx.x & 31;
    const int half = lane >> 4;          // 0 or 1
    const int r    = lane & 15;

    const int tileM = blockIdx.x * 8 + wave;
    if (tileM >= n_tiles) return;        // wave-uniform: EXEC stays all-1s

    // A row for this lane (clamped for a possible partial last tile)
    int mrow   = tileM * 16 + r;
    int mclamp = mrow < n_nodes ? mrow : (n_nodes - 1);
    const float* xrow = X + (size_t)mclamp * DIN;

    v8f acc0 = {}, acc1 = {}, acc2 = {}, acc3 = {};

    for (int k0 = 0; k0 < DIN; k0 += 4) {
        // A fragment: vgpr v holds K = k0 + 2*half + v -> contiguous float2
        const float2 av = *(const float2*)(xrow + k0 + 2 * half);
        v2f a; a.x = av.x; a.y = av.y;

        // B fragments: one aligned 8B LDS load per n-tile, no shuffles
        const float* bbase = &WsB[((k0 >> 2) << 8) + lane * 2];
        v2f b0 = *(const v2f*)(bbase + 0);
        v2f b1 = *(const v2f*)(bbase + 64);
        v2f b2 = *(const v2f*)(bbase + 128);
        v2f b3 = *(const v2f*)(bbase + 192);

        acc0 = __builtin_amdgcn_wmma_f32_16x16x4_f32(false, a, false, b0, (short)0, acc0, false, false);
        acc1 = __builtin_amdgcn_wmma_f32_16x16x4_f32(false, a, false, b1, (short)0, acc1, false, false);
        acc2 = __builtin_amdgcn_wmma_f32_16x16x4_f32(false, a, false, b2, (short)0, acc2, false, false);
        acc3 = __builtin_amdgcn_wmma_f32_16x16x4_f32(false, a, false, b3, (short)0, acc3, false, false);
    }

    // C/D layout: vgpr j -> row M = j + 8*half, col N = r (+16 per n-tile)
    const int mbase = tileM * 16 + 8 * half;
#pragma unroll
    for (int j = 0; j < 8; ++j) {
        int m = mbase + j;
        if (m < n_nodes) {               // lane-predicated store only
            float* orow = support + (size_t)m * DOUT + r;
            orow[0]  = acc0[j];
            orow[16] = acc1[j];
            orow[32] = acc2[j];
            orow[48] = acc3[j];
        }
    }
}

// ---------------------------------------------------------------------------
// Kernel 2: COO scatter  out[rows[e]] += vals[e] * support[cols[e]]
// 64 threads per edge (one per output feature); gather is a 256B coalesced
// read (support fits in the 192MB L2), scatter via native f32 global atomics.
// ---------------------------------------------------------------------------
__global__ void __launch_bounds__(256)
gcn_spmm_scatter(const float* __restrict__ support,
                 const int* __restrict__ rows, const int* __restrict__ cols,
                 const float* __restrict__ vals, float* __restrict__ out,
                 int n_edges) {
    long long t = (long long)blockIdx.x * blockDim.x + threadIdx.x;
    int e = (int)(t >> 6);
    int f = (int)(t & 63);
    if (e >= n_edges) return;

    int   rd = rows[e];
    int   cs = cols[e];
    float v  = vals[e];
    float msg = v * support[(size_t)cs * DOUT + f];
    unsafeAtomicAdd(&out[(size_t)rd * DOUT + f], msg);   // global_atomic_add_f32
}

// ---------------------------------------------------------------------------
extern "C" void kernel_launch(void* const* d_in, const int* in_sizes, int n_in,
                              void* d_out, int out_size, void* d_ws, size_t ws_size,
                              hipStream_t stream) {
    const float* X      = (const float*)d_in[0];
    const float* W      = (const float*)d_in[1];
    const int*   A_rows = (const int*)d_in[2];
    const int*   A_cols = (const int*)d_in[3];
    const float* A_vals = (const float*)d_in[4];
    float*       out    = (float*)d_out;
    float*       support = (float*)d_ws;   // n_nodes * 64 fp32 = 25.6 MB scratch

    const int n_nodes = in_sizes[0] / DIN;
    const int n_edges = in_sizes[2];

    // out must start at zero for the scatter-add (harness poisons it).
    hipMemsetAsync(out, 0, (size_t)out_size * sizeof(float), stream);

    const int n_tiles = (n_nodes + 15) / 16;
    const int gemm_blocks = (n_tiles + 7) / 8;          // 8 waves per block
    gcn_gemm_wmma<<<gemm_blocks, 256, 0, stream>>>(X, W, support, n_nodes, n_tiles);

    const long long sp_threads = (long long)n_edges * 64;
    const int sp_blocks = (int)((sp_threads + 255) / 256);
    gcn_spmm_scatter<<<sp_blocks, 256, 0, stream>>>(support, A_rows, A_cols, A_vals,
                                                    out, n_edges);
}